// DPhysicsEngine_44719199486202
// MI455X (gfx1250) — compile-verified
//
#include <hip/hip_runtime.h>
#include <math.h>

// ---------------- problem constants (match reference) ----------------
#define D_MAXF   6.4f
#define MASSF    40.0f
#define GRAVF    9.8f
#define N_PARTS  4
#define Bn       512
#define NPTS     2048
#define Hg       256
#define Wg       256
#define HW       (Hg * Wg)

typedef __attribute__((ext_vector_type(2))) float v2f;
typedef __attribute__((ext_vector_type(8))) float v8f;

// 4-byte-aligned float pair: lets the backend emit global_load_b64 at dword
// alignment (observed legal for gfx1250 in round-1/2 codegen).
typedef struct __attribute__((packed, aligned(4))) { float x, y; } f2u;

// Hardware tanh (CDNA5 V_TANH_F32 transcendental) when the builtin exists;
// fall back to libm so the compile can never break.
#if __has_builtin(__builtin_amdgcn_tanhf)
  #define TANHF(v) __builtin_amdgcn_tanhf(v)
#elif __has_builtin(__builtin_amdgcn_tanh_f32)
  #define TANHF(v) __builtin_amdgcn_tanh_f32(v)
#else
  #define TANHF(v) tanhf(v)
#endif

// Output layout (flat concat of the reference's return tuple):
//   xd      [B,3]        @ 0
//   xdd     [B,3]        @ B*3
//   omega_d [B,3]        @ 2*B*3
//   F_spring[B,NPTS,3]   @ 3*B*3
//   F_frict [B,NPTS,3]   @ 3*B*3 + B*NPTS*3
#define OFF_XD   ((size_t)0)
#define OFF_XDD  ((size_t)Bn * 3)
#define OFF_OMD  ((size_t)Bn * 6)
#define OFF_FS   ((size_t)Bn * 9)
#define OFF_FF   (OFF_FS + (size_t)Bn * NPTS * 3)

__global__ __launch_bounds__(256)
void DPhysicsEngine_44719199486202_kernel(
    const float* __restrict__ x,          // [B,3]
    const float* __restrict__ xd,         // [B,3]
    const float* __restrict__ R,          // [B,3,3] row-major
    const float* __restrict__ omega,      // [B,3]
    const float* __restrict__ Imat,      // [B,3,3]
    const float* __restrict__ rpts,       // [B,NPTS,3]
    const float* __restrict__ zg,         // [B,H,W]
    const float* __restrict__ zgrad,      // [B,2,H,W]
    const float* __restrict__ kg,         // [B,H,W]
    const float* __restrict__ dg,         // [B,H,W]
    const float* __restrict__ fg,         // [B,H,W]
    const int*   __restrict__ dparts,     // [NPTS]
    const float* __restrict__ ctrl,       // [B,N_PARTS]
    float* __restrict__ out)
{
    __shared__ float s_xpts[NPTS * 3];    // 24 KB: transformed points, later reused for reduction

    const int tid  = threadIdx.x;
    const int lane = tid & 31;
    const int wv   = tid >> 5;            // 8 waves of 32
    const int b    = blockIdx.x;

    // ---- per-body uniforms (uniform addresses -> scalar loads) ----
    const float* Rb = R + (size_t)b * 9;
    const float x0 = x[b*3+0], x1 = x[b*3+1], x2 = x[b*3+2];
    const float xd0 = xd[b*3+0], xd1 = xd[b*3+1], xd2 = xd[b*3+2];
    const float om0 = omega[b*3+0], om1 = omega[b*3+1], om2 = omega[b*3+2];

    // thrust_dir = normalize(R[:,0])
    float tcx = Rb[0], tcy = Rb[3], tcz = Rb[6];
    {
        float inv = __frsqrt_rn(tcx*tcx + tcy*tcy + tcz*tcz);
        tcx *= inv; tcy *= inv; tcz *= inv;
    }

    // =====================================================================
    // Phase A: affine transform of 2048 points via V_WMMA_F32_16X16X4_F32.
    //   D[m][n] = sum_k A[m][k] * B[k][n],  A[m] = (px,py,pz,1),
    //   B[k][n] = R[n][k] (k<3), B[3][n] = x[n]  (cols n>=3 are zero)
    // A layout (f32 16x4): lanes 0-15 hold {K0,K1}, lanes 16-31 hold {K2,K3}.
    // B layout (f32 4x16): vgpr0 = rows K0|K1 on lane halves, vgpr1 = K2|K3.
    // D layout (f32 16x16): vgpr r -> M=r (lanes 0-15, N=lane) / M=r+8 (lanes 16-31).
    // Software-pipelined in groups of 4: issue 4 b64 loads, then consume.
    // =====================================================================
    const bool lo   = lane < 16;
    const int  nsel = lane & 15;
    v2f bop; bop.x = 0.0f; bop.y = 0.0f;
    if (nsel < 3) {
        if (lo) { bop.x = Rb[nsel*3 + 0]; bop.y = Rb[nsel*3 + 2]; }
        else    { bop.x = Rb[nsel*3 + 1]; bop.y = x[b*3 + nsel]; }
    }

    const float* P0 = rpts + (size_t)b * NPTS * 3 + (lo ? 0 : 1);
    for (int tt = 0; tt < 4; ++tt) {
        f2u pr[4];
        #pragma unroll
        for (int u = 0; u < 4; ++u) {
            const int base = wv * 256 + (tt * 4 + u) * 16;
            pr[u] = *(const f2u*)(P0 + (size_t)(base + nsel) * 3);
        }
        #pragma unroll
        for (int u = 0; u < 4; ++u) {
            const int base = wv * 256 + (tt * 4 + u) * 16;
            v2f aop;
            aop.x = lo ? pr[u].x : pr[u].y;   // px | pz
            aop.y = lo ? pr[u].y : 1.0f;      // py | homogeneous 1
            v8f acc = {};
            v8f d = __builtin_amdgcn_wmma_f32_16x16x4_f32(
                        false, aop, false, bop, (short)0, acc, false, false);
            if (nsel < 3) {                   // N index = coordinate
                const int mb = base + (lo ? 0 : 8);
                #pragma unroll
                for (int r = 0; r < 8; ++r)
                    s_xpts[(mb + r) * 3 + nsel] = d[r];
            }
        }
    }
    __syncthreads();

    // =====================================================================
    // Phase B: per-point contact forces. 8 points per thread, stride 256.
    // =====================================================================
    float fs0=0.f,fs1=0.f,fs2=0.f, ff0=0.f,ff1=0.f,ff2=0.f, tq0=0.f,tq1=0.f,tq2=0.f;

    const float SC = (float)(Hg - 1) / (2.0f * D_MAXF);
    const size_t gbase = (size_t)b * HW;
    const float* zgb = zg + gbase;
    const float* kgb = kg + gbase;
    const float* dgb = dg + gbase;
    const float* fgb = fg + gbase;
    const float* gxb = zgrad + (size_t)b * 2 * HW;
    const float* gyb = gxb + HW;
    float* __restrict__ oFs = out + OFF_FS + (size_t)b * NPTS * 3;
    float* __restrict__ oFf = out + OFF_FF + (size_t)b * NPTS * 3;

    #pragma unroll 2
    for (int k = 0; k < 8; ++k) {
        const int n = tid + (k << 8);
        const float px = s_xpts[n*3+0];
        const float py = s_xpts[n*3+1];
        const float pz = s_xpts[n*3+2];

        // shared bilinear setup for all 6 grids
        float gi = fminf(fmaxf((px + D_MAXF) * SC, 0.0f), (float)(Hg - 1));
        float gj = fminf(fmaxf((py + D_MAXF) * SC, 0.0f), (float)(Wg - 1));
        int i0 = (int)floorf(gi); i0 = (i0 > Hg - 2) ? Hg - 2 : i0;
        int j0 = (int)floorf(gj); j0 = (j0 > Wg - 2) ? Wg - 2 : j0;
        const float wi = gi - (float)i0, wj = gj - (float)j0;
        const float w00 = (1.f-wi)*(1.f-wj), w01 = (1.f-wi)*wj;
        const float w10 = wi*(1.f-wj),       w11 = wi*wj;
        const int off = i0 * Wg + j0;

        // each grid: two dword-aligned b64 loads (top row pair, bottom row pair)
        auto BIL = [&](const float* __restrict__ g) -> float {
            const f2u tp = *(const f2u*)(g + off);
            const f2u bt = *(const f2u*)(g + off + Wg);
            return tp.x*w00 + tp.y*w01 + bt.x*w10 + bt.y*w11;
        };
        const float zv  = BIL(zgb);
        const float kv  = BIL(kgb);
        const float dmp = BIL(dgb);
        const float muv = BIL(fgb);
        const float gx  = BIL(gxb);
        const float gy  = BIL(gyb);

        const float dh = pz - zv;
        const bool ong = (px >= -D_MAXF) & (px <= D_MAXF) & (py >= -D_MAXF) & (py <= D_MAXF);
        const float contact = ((dh <= 0.0f) & ong) ? 1.0f : 0.0f;

        // surface normal (hw v_rsq_f32)
        const float inrm = __frsqrt_rn(gx*gx + gy*gy + 1.0f);
        const float nx = -gx*inrm, ny = -gy*inrm, nz = inrm;

        // point velocity: xd + omega x r,  r = x_point - x
        const float rx = px - x0, ry = py - x1, rz = pz - x2;
        const float vx = xd0 + (om1*rz - om2*ry);
        const float vy = xd1 + (om2*rx - om0*rz);
        const float vz = xd2 + (om0*ry - om1*rx);
        const float vn = vx*nx + vy*ny + vz*nz;

        // spring force (already masked & /n_pts)
        const float coef = -(kv*dh + dmp*vn) * contact * (1.0f / (float)NPTS);
        const float Fsx = coef*nx, Fsy = coef*ny, Fsz = coef*nz;
        const float Nmag = fabsf(coef);

        // friction (only the point's driving part matters)
        const int part = dparts[n];
        const float cm = ctrl[b*N_PARTS + part];
        const float dvx = cm*tcx - vx, dvy = cm*tcy - vy, dvz = cm*tcz - vz;
        const float dvn = dvx*nx + dvy*ny + dvz*nz;
        const float mn  = muv * Nmag;
        const float Ffx = mn * TANHF(dvx - dvn*nx);
        const float Ffy = mn * TANHF(dvy - dvn*ny);
        const float Ffz = mn * TANHF(dvz - dvn*nz);

        const size_t o = (size_t)n * 3;
        oFs[o+0] = Fsx; oFs[o+1] = Fsy; oFs[o+2] = Fsz;
        oFf[o+0] = Ffx; oFf[o+1] = Ffy; oFf[o+2] = Ffz;

        const float Ftx = Fsx + Ffx, Fty = Fsy + Ffy, Ftz = Fsz + Ffz;
        tq0 += ry*Ftz - rz*Fty;
        tq1 += rz*Ftx - rx*Ftz;
        tq2 += rx*Fty - ry*Ftx;
        fs0 += Fsx; fs1 += Fsy; fs2 += Fsz;
        ff0 += Ffx; ff1 += Ffy; ff2 += Ffz;
    }

    // =====================================================================
    // Phase C: block reduction (reuse s_xpts) + finalize
    // =====================================================================
    __syncthreads();
    float* red = s_xpts;                  // needs 9*256 = 2304 floats <= 6144
    red[0*256+tid]=fs0; red[1*256+tid]=fs1; red[2*256+tid]=fs2;
    red[3*256+tid]=ff0; red[4*256+tid]=ff1; red[5*256+tid]=ff2;
    red[6*256+tid]=tq0; red[7*256+tid]=tq1; red[8*256+tid]=tq2;
    __syncthreads();
    for (int s = 128; s > 0; s >>= 1) {
        if (tid < s) {
            #pragma unroll
            for (int q = 0; q < 9; ++q)
                red[q*256+tid] += red[q*256+tid+s];
        }
        __syncthreads();
    }

    if (tid < 3) {
        const int c = tid;
        // pass-through xd
        out[OFF_XD + (size_t)b*3 + c] = xd[b*3 + c];
        // xdd = (F_grav + sum Fs + sum Ff) / MASS
        float Fc = red[c*256] + red[(3+c)*256] + ((c == 2) ? (-MASSF * GRAVF) : 0.0f);
        out[OFF_XDD + (size_t)b*3 + c] = Fc / MASSF;
        // omega_d = I^-1 * torque  (adjugate / det; inv[i][j] = C[j][i]/det)
        const float* Ib = Imat + (size_t)b * 9;
        const float a00=Ib[0],a01=Ib[1],a02=Ib[2],
                    a10=Ib[3],a11=Ib[4],a12=Ib[5],
                    a20=Ib[6],a21=Ib[7],a22=Ib[8];
        const float C00 =  (a11*a22 - a12*a21);
        const float C01 = -(a10*a22 - a12*a20);
        const float C02 =  (a10*a21 - a11*a20);
        const float C10 = -(a01*a22 - a02*a21);
        const float C11 =  (a00*a22 - a02*a20);
        const float C12 = -(a00*a21 - a01*a20);
        const float C20 =  (a01*a12 - a02*a11);
        const float C21 = -(a00*a12 - a02*a10);
        const float C22 =  (a00*a11 - a01*a10);
        const float invdet = 1.0f / (a00*C00 + a01*C01 + a02*C02);
        const float t0 = red[6*256], t1 = red[7*256], t2 = red[8*256];
        float od;
        if      (c == 0) od = (C00*t0 + C10*t1 + C20*t2) * invdet;
        else if (c == 1) od = (C01*t0 + C11*t1 + C21*t2) * invdet;
        else             od = (C02*t0 + C12*t1 + C22*t2) * invdet;
        out[OFF_OMD + (size_t)b*3 + c] = od;
    }
}

extern "C" void kernel_launch(void* const* d_in, const int* in_sizes, int n_in,
                              void* d_out, int out_size, void* d_ws, size_t ws_size,
                              hipStream_t stream) {
    (void)in_sizes; (void)n_in; (void)out_size; (void)d_ws; (void)ws_size;
    const float* x      = (const float*)d_in[0];
    const float* xd     = (const float*)d_in[1];
    const float* R      = (const float*)d_in[2];
    const float* omega  = (const float*)d_in[3];
    const float* Imat   = (const float*)d_in[4];
    const float* rpts   = (const float*)d_in[5];
    const float* zg     = (const float*)d_in[6];
    const float* zgrad  = (const float*)d_in[7];
    const float* kg     = (const float*)d_in[8];
    const float* dg     = (const float*)d_in[9];
    const float* fg     = (const float*)d_in[10];
    const int*   dparts = (const int*)d_in[11];
    const float* ctrl   = (const float*)d_in[12];
    float* out = (float*)d_out;

    DPhysicsEngine_44719199486202_kernel<<<Bn, 256, 0, stream>>>(
        x, xd, R, omega, Imat, rpts, zg, zgrad, kg, dg, fg, dparts, ctrl, out);
}